// MaskedSelfAttention_52733608461098
// MI455X (gfx1250) — compile-verified
//
#include <hip/hip_runtime.h>

typedef __attribute__((ext_vector_type(2))) float v2f;
typedef __attribute__((ext_vector_type(8))) float v8f;
typedef __attribute__((ext_vector_type(4))) unsigned int u32x4;
typedef __attribute__((ext_vector_type(4))) int i32x4;
typedef __attribute__((ext_vector_type(8))) int i32x8;

#define N_PIX 4096
#define C_IN  64
#define CQ    8
#define NEGV  (-1e9f)

#define LDS_ROW 18                 // 16 data dwords + 2 pad dwords (bank-conflict free)
#define LDS_BUF (C_IN * LDS_ROW)   // 1152 dwords per buffer

#if defined(__has_builtin)
#if __has_builtin(__builtin_amdgcn_tensor_load_to_lds) && __has_builtin(__builtin_amdgcn_s_wait_tensorcnt)
#define HAS_TDM 1
#endif
#endif
#ifndef HAS_TDM
#define HAS_TDM 0
#endif

// ---------------------------------------------------------------------------
// Kernel 1: fused 1x1-conv projections -> q,k (B,N,8), v (B,C,N).
// ---------------------------------------------------------------------------
__global__ __launch_bounds__(256) void proj_kernel(
    const float* __restrict__ x,
    const float* __restrict__ Wq, const float* __restrict__ bq,
    const float* __restrict__ Wk, const float* __restrict__ bk,
    const float* __restrict__ Wv, const float* __restrict__ bv,
    float* __restrict__ qws, float* __restrict__ kws, float* __restrict__ vws)
{
    const int idx = blockIdx.x * 256 + threadIdx.x;   // global pixel id
    const int b = idx >> 12;
    const int n = idx & (N_PIX - 1);
    const float* xp = x + (size_t)b * C_IN * N_PIX + n;

    float aq[CQ], ak[CQ], av[C_IN];
#pragma unroll
    for (int o = 0; o < CQ; ++o) { aq[o] = bq[o]; ak[o] = bk[o]; }
#pragma unroll
    for (int o = 0; o < C_IN; ++o) av[o] = bv[o];

    for (int c = 0; c < C_IN; ++c) {
        const float xv = xp[(size_t)c * N_PIX];
#pragma unroll
        for (int o = 0; o < CQ; ++o) {
            aq[o] = fmaf(Wq[o * C_IN + c], xv, aq[o]);
            ak[o] = fmaf(Wk[o * C_IN + c], xv, ak[o]);
        }
#pragma unroll
        for (int o = 0; o < C_IN; ++o)
            av[o] = fmaf(Wv[o * C_IN + c], xv, av[o]);
    }

    float* qp = qws + (size_t)idx * CQ;
    float* kp = kws + (size_t)idx * CQ;
#pragma unroll
    for (int o = 0; o < CQ; ++o) { qp[o] = aq[o]; kp[o] = ak[o]; }
    float* vp = vws + (size_t)b * C_IN * N_PIX + n;
#pragma unroll
    for (int o = 0; o < C_IN; ++o) vp[(size_t)o * N_PIX] = av[o];
}

// ---------------------------------------------------------------------------
// TDM: DMA one v-tile (2D: 16 pixels x 64 channels, row stride 4096 floats)
// into LDS, padding each 16-dword row with 2 dwords (LDS row stride 18).
// D# bitfields per CDNA5 ISA ch.8: group0 {count,lds_addr,global_addr,type=2},
// group1 {data_size=4B, pad_en, pad_interval=16dw, pad_amount=2dw, dims}.
// ---------------------------------------------------------------------------
#if HAS_TDM
__device__ __forceinline__ void tdm_load_vtile(const float* gsrc, unsigned lds_off) {
    const unsigned long long ga = (unsigned long long)(uintptr_t)gsrc;
    u32x4 g0;
    g0[0] = 1u;                                               // count=1, user mode
    g0[1] = lds_off;                                          // lds_addr (bytes)
    g0[2] = (unsigned)(ga & 0xFFFFFFFFu);                     // global_addr[31:0]
    g0[3] = (unsigned)((ga >> 32) & 0x01FFFFFFu) | (2u << 30);// global_addr[56:32]|type=2
    i32x8 g1;
    g1[0] = (int)0x02D20000;      // data_size=2(4B), pad_en, pad_interval=3(16dw), pad_amount=1(2dw)
    g1[1] = (int)(N_PIX << 16);   // tensor_dim0[15:0] << 16 (=4096)
    g1[2] = (int)(C_IN << 16);    // tensor_dim0 hi=0 | tensor_dim1[15:0]=64
    g1[3] = (int)(16 << 16);      // tensor_dim1 hi=0 | tile_dim0=16
    g1[4] = (int)C_IN;            // tile_dim1=64, tile_dim2=0
    g1[5] = (int)N_PIX;           // tensor_dim0_stride lo32 = 4096
    g1[6] = 0;                    // stride0 hi | tensor_dim1_stride lo
    g1[7] = 0;
    i32x4 gz = {0, 0, 0, 0};
#if __clang_major__ >= 23
    i32x8 gz8 = {0, 0, 0, 0, 0, 0, 0, 0};
    __builtin_amdgcn_tensor_load_to_lds(g0, g1, gz, gz, gz8, 0);
#else
    __builtin_amdgcn_tensor_load_to_lds(g0, g1, gz, gz, 0);
#endif
}
#endif

// ---------------------------------------------------------------------------
// Kernel 2: flash attention, one wave32 per 16-query tile, 8 waves/block.
// All waves in a block share batch b and the key-tile sequence, so wave 0
// TDM-prefetches each 4KB v-tile into a double-buffered LDS staging area.
// ---------------------------------------------------------------------------
__global__ __launch_bounds__(256) void attn_kernel(
    const float* __restrict__ x, const int* __restrict__ mask,
    const float* __restrict__ qws, const float* __restrict__ kws,
    const float* __restrict__ vws, const float* __restrict__ gammaP,
    float* __restrict__ out)
{
    const int lane = threadIdx.x & 31;
    const int wv   = threadIdx.x >> 5;
    const int tile = blockIdx.x * 8 + wv;       // 0..1023
    const int b    = tile >> 8;                 // block-uniform (32 blocks/batch)
    const int i0   = (tile & 255) << 4;
    const int h    = lane >> 4;
    const int lm   = lane & 15;

#if HAS_TDM
    __shared__ float vlds[2][LDS_BUF];
#endif

    const float gamma = gammaP[0];

    // Q as B-fragments (K = vgpr + 2h)
    const float* qp = qws + ((size_t)b * N_PIX + i0 + lm) * CQ;
    const v2f qb0 = *(const v2f*)(qp + 2 * h);
    const v2f qb1 = *(const v2f*)(qp + 4 + 2 * h);

    // query validity bit (identical for both halves)
    const unsigned qbal = (unsigned)__ballot(mask[b * N_PIX + i0 + lm] != 0);
    const unsigned qv   = (qbal >> lm) & 1u;

    v8f acc0 = {}, acc1 = {}, acc2 = {}, acc3 = {};
    float m = -INFINITY;
    float l = 0.0f;

    const float* kbase = kws + (size_t)b * N_PIX * CQ;
    const float* vbase = vws + (size_t)b * C_IN * N_PIX;
    const int*   mbase = mask + b * N_PIX;

#if HAS_TDM
    if (wv == 0) {   // wave-uniform: prime both staging buffers
        tdm_load_vtile(vbase + 0,  (unsigned)(uintptr_t)&vlds[0][0]);
        tdm_load_vtile(vbase + 16, (unsigned)(uintptr_t)&vlds[1][0]);
    }
#endif

    for (int t = 0; t < N_PIX / 16; ++t) {
        const int j0 = t << 4;

#if HAS_TDM
        if (wv == 0) {  // TDM completes in order: cnt<=1 means tile t landed
            if (t == N_PIX / 16 - 1) __builtin_amdgcn_s_wait_tensorcnt(0);
            else                     __builtin_amdgcn_s_wait_tensorcnt(1);
        }
        __syncthreads();
        const float* vtile = &vlds[t & 1][0];
#endif

        // K A-fragments
        const float* kp = kbase + (size_t)(j0 + lm) * CQ;
        const v2f ka0 = *(const v2f*)(kp + 2 * h);
        const v2f ka1 = *(const v2f*)(kp + 4 + 2 * h);

        unsigned kbal = (unsigned)__ballot(mbase[j0 + lm] != 0);
        kbal = qv ? kbal : 0u;               // fold query validity
        const unsigned kb = kbal >> (8 * h); // per-half key bits

        // S^T = K_tile x Q^T  (two K=4 chunks)
        v8f s = {};
        s = __builtin_amdgcn_wmma_f32_16x16x4_f32(false, ka0, false, qb0,
                                                  (short)0, s, false, false);
        s = __builtin_amdgcn_wmma_f32_16x16x4_f32(false, ka1, false, qb1,
                                                  (short)0, s, false, false);

        // mask + online softmax stats
        float p[8];
        float tmax = -INFINITY;
#pragma unroll
        for (int r = 0; r < 8; ++r) {
            const float sv = (kb & (1u << r)) ? s[r] : NEGV;
            p[r] = sv;
            tmax = fmaxf(tmax, sv);
        }
        tmax = fmaxf(tmax, __shfl_xor(tmax, 16));
        const float mnew  = fmaxf(m, tmax);
        const float scale = __expf(m - mnew);
        float psum = 0.0f;
#pragma unroll
        for (int r = 0; r < 8; ++r) {
            p[r] = __expf(p[r] - mnew);
            psum += p[r];
        }
        psum += __shfl_xor(psum, 16);
        l = l * scale + psum;
        m = mnew;
#pragma unroll
        for (int r = 0; r < 8; ++r) {
            acc0[r] *= scale; acc1[r] *= scale;
            acc2[r] *= scale; acc3[r] *= scale;
        }

        // Re-lay P into B fragments (key = 4kc + v + 2h at query lm)
        float pl[8], ph[8];
#pragma unroll
        for (int r = 0; r < 8; ++r) {
            pl[r] = __shfl(p[r], lm);
            ph[r] = __shfl(p[r], lm + 16);
        }
        v2f bp[4];
#pragma unroll
        for (int kc = 0; kc < 4; ++kc) {
            const int r0 = (kc & 1) * 4;
            const float s0 = (kc < 2) ? pl[r0 + 0] : ph[r0 + 0];
            const float s1 = (kc < 2) ? pl[r0 + 1] : ph[r0 + 1];
            const float s2 = (kc < 2) ? pl[r0 + 2] : ph[r0 + 2];
            const float s3 = (kc < 2) ? pl[r0 + 3] : ph[r0 + 3];
            bp[kc].x = h ? s2 : s0;
            bp[kc].y = h ? s3 : s1;
        }

        // V A-fragments: channel = cg*16+lm, keys j0+4kc+2h..+1
        v2f va[4][4];
#pragma unroll
        for (int cg = 0; cg < 4; ++cg) {
#if HAS_TDM
            const float* vp = vtile + (cg * 16 + lm) * LDS_ROW + 2 * h;
#else
            const float* vp = vbase + (size_t)(cg * 16 + lm) * N_PIX + j0 + 2 * h;
#endif
#pragma unroll
            for (int kc = 0; kc < 4; ++kc)
                va[cg][kc] = *(const v2f*)(vp + 4 * kc);
        }

        // out_tile += V_tile x P^T
#pragma unroll
        for (int kc = 0; kc < 4; ++kc) {
            acc0 = __builtin_amdgcn_wmma_f32_16x16x4_f32(false, va[0][kc], false, bp[kc],
                                                         (short)0, acc0, false, false);
            acc1 = __builtin_amdgcn_wmma_f32_16x16x4_f32(false, va[1][kc], false, bp[kc],
                                                         (short)0, acc1, false, false);
            acc2 = __builtin_amdgcn_wmma_f32_16x16x4_f32(false, va[2][kc], false, bp[kc],
                                                         (short)0, acc2, false, false);
            acc3 = __builtin_amdgcn_wmma_f32_16x16x4_f32(false, va[3][kc], false, bp[kc],
                                                         (short)0, acc3, false, false);
        }

#if HAS_TDM
        __syncthreads();  // everyone done reading buffer (t&1)
        if (wv == 0 && t + 2 < N_PIX / 16)
            tdm_load_vtile(vbase + (j0 + 32), (unsigned)(uintptr_t)&vlds[t & 1][0]);
#endif
    }

    // epilogue: out = gamma * acc/l + x   (c = cg*16 + r + 8h, n = i0+lm)
    const float rl = 1.0f / l;
    const size_t obase = (size_t)b * C_IN * N_PIX + i0 + lm;
#pragma unroll
    for (int cg = 0; cg < 4; ++cg) {
        const v8f a = (cg == 0) ? acc0 : (cg == 1) ? acc1 : (cg == 2) ? acc2 : acc3;
#pragma unroll
        for (int r = 0; r < 8; ++r) {
            const int c = cg * 16 + r + 8 * h;
            const size_t idx = obase + (size_t)c * N_PIX;
            out[idx] = fmaf(gamma, a[r] * rl, x[idx]);
        }
    }
}

// ---------------------------------------------------------------------------
extern "C" void kernel_launch(void* const* d_in, const int* in_sizes, int n_in,
                              void* d_out, int out_size, void* d_ws, size_t ws_size,
                              hipStream_t stream) {
    const float* x     = (const float*)d_in[0];
    const int*   mask  = (const int*)d_in[1];
    const float* Wq    = (const float*)d_in[2];
    const float* bq    = (const float*)d_in[3];
    const float* Wk    = (const float*)d_in[4];
    const float* bk    = (const float*)d_in[5];
    const float* Wv    = (const float*)d_in[6];
    const float* bv    = (const float*)d_in[7];
    const float* gamma = (const float*)d_in[8];
    float* out = (float*)d_out;

    float* qws = (float*)d_ws;                    // 4*4096*8  floats
    float* kws = qws + 4 * N_PIX * CQ;            // 4*4096*8  floats
    float* vws = kws + 4 * N_PIX * CQ;            // 4*64*4096 floats (5 MB total)

    proj_kernel<<<64, 256, 0, stream>>>(x, Wq, bq, Wk, bk, Wv, bv, qws, kws, vws);
    attn_kernel<<<128, 256, 0, stream>>>(x, mask, qws, kws, vws, gamma, out);
}